// Biquad_54571854463265
// MI455X (gfx1250) — compile-verified
//
#include <hip/hip_runtime.h>
#include <stdint.h>

#define BATCH    64
#define SAMPLES  524288
#define CL       4096                 // samples per chunk (one wave per chunk)
#define SUBL     128                  // samples per lane
#define NCHUNK   (SAMPLES / CL)       // 128 chunks per row
#define LSTRIDE  132                  // padded floats per lane region (16B aligned)
#define NASYNC   (CL / (32 * 4))      // 32 b128 async issues per chunk

// ---------------- device helpers ----------------

// Cooperative, fully coalesced async DMA of one 4096-sample chunk into padded LDS.
// LAST_USE=0: regular-temporal load (populate L2; pass 1).
// LAST_USE=1: last-use load (discard L2 line after hit; pass 2).
template <int LAST_USE>
__device__ __forceinline__ void async_load_chunk(const float* __restrict__ gsrc,
                                                 float* xb, int lane) {
  #pragma unroll 4
  for (int n = 0; n < NASYNC; ++n) {
    int fidx = (n * 32 + lane) * 4;          // float index within chunk (x4 aligned)
    int i = fidx >> 7;                       // destination lane-region
    int t = fidx & (SUBL - 1);
    uint32_t ldsb = (uint32_t)(uintptr_t)(&xb[i * LSTRIDE + t]);
    const float* g = gsrc + fidx;
    if (LAST_USE) {
      asm volatile("global_load_async_to_lds_b128 %0, %1, off th:TH_LOAD_LU"
                   :: "v"(ldsb), "v"(g) : "memory");
    } else {
      asm volatile("global_load_async_to_lds_b128 %0, %1, off"
                   :: "v"(ldsb), "v"(g) : "memory");
    }
  }
  asm volatile("s_wait_asynccnt 0x0" ::: "memory");
}

// Run the biquad recurrence over this lane's 128-sample sub-chunk from zero state;
// result (z1,z2) is the lane's forced-response vector.
__device__ __forceinline__ void local_forced(const float* xb, int lane,
                                             float b0, float b1, float b2,
                                             float a1, float a2,
                                             float& f1, float& f2) {
  float z1 = 0.f, z2 = 0.f;
  const float* p = xb + lane * LSTRIDE;
  #pragma unroll 4
  for (int t = 0; t < SUBL; t += 4) {
    float4 v = *(const float4*)(p + t);      // ds_load_b128
    float xs0 = v.x, xs1 = v.y, xs2 = v.z, xs3 = v.w;
    float y;
    y = b0*xs0 + z1; { float nz1 = b1*xs0 - a1*y + z2; z2 = b2*xs0 - a2*y; z1 = nz1; }
    y = b0*xs1 + z1; { float nz1 = b1*xs1 - a1*y + z2; z2 = b2*xs1 - a2*y; z1 = nz1; }
    y = b0*xs2 + z1; { float nz1 = b1*xs2 - a1*y + z2; z2 = b2*xs2 - a2*y; z1 = nz1; }
    y = b0*xs3 + z1; { float nz1 = b1*xs3 - a1*y + z2; z2 = b2*xs3 - a2*y; z1 = nz1; }
  }
  f1 = z1; f2 = z2;
}

// Kogge-Stone inclusive scan of affine state vectors across the wave.
// Multiplier at step k is the (row-uniform) matrix A^(SUBL * 2^k) = mrow[4k .. 4k+3].
__device__ __forceinline__ void wave_scan(float& f1, float& f2,
                                          const float* __restrict__ mrow, int lane) {
  #pragma unroll
  for (int k = 0; k < 5; ++k) {
    float p1 = __shfl_up(f1, 1u << k, 32);
    float p2 = __shfl_up(f2, 1u << k, 32);
    float m00 = mrow[4*k+0], m01 = mrow[4*k+1];
    float m10 = mrow[4*k+2], m11 = mrow[4*k+3];
    if (lane >= (1 << k)) {
      float n1 = m00 * p1 + m01 * p2 + f1;
      float n2 = m10 * p1 + m11 * p2 + f2;
      f1 = n1; f2 = n2;
    }
  }
}

// ---------------- kernels ----------------

// One thread per row: A = [[-a1,1],[-a2,0]]; repeated squaring gives
// A^(2^7..2^11) (scan multipliers) and A^(2^12) (inter-chunk multiplier).
__global__ void k_setup(const float* __restrict__ b, const float* __restrict__ a,
                        float* __restrict__ mats) {
  int r = threadIdx.x;
  if (r >= BATCH) return;
  float a1 = a[r*2 + 0], a2 = a[r*2 + 1];
  float p00 = -a1, p01 = 1.f, p10 = -a2, p11 = 0.f;
  float* m = mats + r * 24;
  #pragma unroll
  for (int j = 0; j < 7; ++j) {     // -> A^128
    float q00 = p00*p00 + p01*p10, q01 = p00*p01 + p01*p11;
    float q10 = p10*p00 + p11*p10, q11 = p10*p01 + p11*p11;
    p00 = q00; p01 = q01; p10 = q10; p11 = q11;
  }
  m[0] = p00; m[1] = p01; m[2] = p10; m[3] = p11;
  #pragma unroll
  for (int k = 1; k <= 5; ++k) {    // A^256 .. A^2048 (k=1..4), A^4096 at slot 5
    float q00 = p00*p00 + p01*p10, q01 = p00*p01 + p01*p11;
    float q10 = p10*p00 + p11*p10, q11 = p10*p01 + p11*p11;
    p00 = q00; p01 = q01; p10 = q10; p11 = q11;
    m[4*k+0] = p00; m[4*k+1] = p01; m[4*k+2] = p10; m[4*k+3] = p11;
  }
}

// Pass 1: per-chunk forced response via lane recurrence + wave scan.
__global__ __launch_bounds__(32)
void k_pass1(const float* __restrict__ x, const float* __restrict__ b,
             const float* __restrict__ a, const float* __restrict__ mats,
             float* __restrict__ fch) {
  __shared__ float xb[32 * LSTRIDE];
  int chunk = blockIdx.x, row = blockIdx.y, lane = threadIdx.x;
  const float* gsrc = x + (size_t)row * SAMPLES + (size_t)chunk * CL;
  async_load_chunk<0>(gsrc, xb, lane);             // RT: populate L2 for pass 2
  float b0 = b[row*3+0], b1 = b[row*3+1], b2 = b[row*3+2];
  float a1 = a[row*2+0], a2 = a[row*2+1];
  float f1, f2;
  local_forced(xb, lane, b0, b1, b2, a1, a2, f1, f2);
  wave_scan(f1, f2, mats + row * 24, lane);
  if (lane == 31) {
    fch[((size_t)row * NCHUNK + chunk) * 2 + 0] = f1;
    fch[((size_t)row * NCHUNK + chunk) * 2 + 1] = f2;
  }
}

// Inter-chunk sequential affine scan: one thread per row, 128 steps.
__global__ void k_cscan(const float* __restrict__ mats, const float* __restrict__ fch,
                        float* __restrict__ sst) {
  int r = threadIdx.x;
  if (r >= BATCH) return;
  const float* m = mats + r * 24 + 20;            // A^4096
  float m00 = m[0], m01 = m[1], m10 = m[2], m11 = m[3];
  float s1 = 0.f, s2 = 0.f;
  for (int j = 0; j < NCHUNK; ++j) {
    size_t idx = ((size_t)r * NCHUNK + j) * 2;
    sst[idx + 0] = s1; sst[idx + 1] = s2;         // start state of chunk j
    float f1 = fch[idx + 0], f2 = fch[idx + 1];
    float n1 = m00 * s1 + m01 * s2 + f1;
    float n2 = m10 * s1 + m11 * s2 + f2;
    s1 = n1; s2 = n2;
  }
}

// Pass 2: exact outputs. Re-scan with chunk start state injected into lane 0,
// shuffle to get each lane's initial state, emit y into LDS, async DMA out.
__global__ __launch_bounds__(32)
void k_pass2(const float* __restrict__ x, const float* __restrict__ b,
             const float* __restrict__ a, const float* __restrict__ mats,
             const float* __restrict__ sst, float* __restrict__ out) {
  __shared__ float xb[32 * LSTRIDE];
  int chunk = blockIdx.x, row = blockIdx.y, lane = threadIdx.x;
  const float* gsrc = x + (size_t)row * SAMPLES + (size_t)chunk * CL;
  async_load_chunk<1>(gsrc, xb, lane);             // LU: drop x lines after this read
  float b0 = b[row*3+0], b1 = b[row*3+1], b2 = b[row*3+2];
  float a1 = a[row*2+0], a2 = a[row*2+1];
  const float* mrow = mats + row * 24;
  size_t cidx = ((size_t)row * NCHUNK + chunk) * 2;
  float s1c = sst[cidx + 0], s2c = sst[cidx + 1];

  float f1, f2;
  local_forced(xb, lane, b0, b1, b2, a1, a2, f1, f2);
  if (lane == 0) {                                 // inject chunk start state
    float n1 = mrow[0] * s1c + mrow[1] * s2c + f1; // M_0 = A^128
    float n2 = mrow[2] * s1c + mrow[3] * s2c + f2;
    f1 = n1; f2 = n2;
  }
  wave_scan(f1, f2, mrow, lane);                   // f = state at END of each lane
  float p1 = __shfl_up(f1, 1, 32);
  float p2 = __shfl_up(f2, 1, 32);
  float z1 = (lane == 0) ? s1c : p1;               // this lane's exact initial state
  float z2 = (lane == 0) ? s2c : p2;

  float* p = xb + lane * LSTRIDE;
  #pragma unroll 4
  for (int t = 0; t < SUBL; t += 4) {
    float4 v = *(float4*)(p + t);
    float y0 = b0*v.x + z1; { float nz1 = b1*v.x - a1*y0 + z2; z2 = b2*v.x - a2*y0; z1 = nz1; }
    float y1 = b0*v.y + z1; { float nz1 = b1*v.y - a1*y1 + z2; z2 = b2*v.y - a2*y1; z1 = nz1; }
    float y2 = b0*v.z + z1; { float nz1 = b1*v.z - a1*y2 + z2; z2 = b2*v.z - a2*y2; z1 = nz1; }
    float y3 = b0*v.w + z1; { float nz1 = b1*v.w - a1*y3 + z2; z2 = b2*v.w - a2*y3; z1 = nz1; }
    v.x = y0; v.y = y1; v.z = y2; v.w = y3;
    *(float4*)(p + t) = v;                         // ds_store_b128 (overwrite x slot)
  }
  asm volatile("s_wait_dscnt 0x0" ::: "memory");   // LDS writes visible to async engine

  float* gdst = out + (size_t)row * SAMPLES + (size_t)chunk * CL;
  #pragma unroll 4
  for (int n = 0; n < NASYNC; ++n) {
    int fidx = (n * 32 + lane) * 4;
    int i = fidx >> 7;
    int t = fidx & (SUBL - 1);
    uint32_t ldsb = (uint32_t)(uintptr_t)(&xb[i * LSTRIDE + t]);
    float* g = gdst + fidx;
    asm volatile("global_store_async_from_lds_b128 %0, %1, off th:TH_STORE_NT"
                 :: "v"(g), "v"(ldsb) : "memory");
  }
  asm volatile("s_wait_asynccnt 0x0" ::: "memory");
}

// ---------------- host launcher ----------------

extern "C" void kernel_launch(void* const* d_in, const int* in_sizes, int n_in,
                              void* d_out, int out_size, void* d_ws, size_t ws_size,
                              hipStream_t stream) {
  const float* x = (const float*)d_in[0];
  const float* b = (const float*)d_in[1];
  const float* a = (const float*)d_in[2];
  float* out = (float*)d_out;

  float* ws   = (float*)d_ws;
  float* mats = ws;                              // BATCH * 24 floats
  float* fch  = mats + (size_t)BATCH * 24;       // BATCH * NCHUNK * 2
  float* sst  = fch  + (size_t)BATCH * NCHUNK * 2;

  k_setup<<<1, BATCH, 0, stream>>>(b, a, mats);
  k_pass1<<<dim3(NCHUNK, BATCH), 32, 0, stream>>>(x, b, a, mats, fch);
  k_cscan<<<1, BATCH, 0, stream>>>(mats, fch, sst);
  k_pass2<<<dim3(NCHUNK, BATCH), 32, 0, stream>>>(x, b, a, mats, sst, out);
}